// VendorGraphSAGE_56633438765477
// MI455X (gfx1250) — compile-verified
//
#include <hip/hip_runtime.h>
#include <math.h>

// GraphSAGE (2x SAGEConv + linear + sigmoid), N=100000, E=1000000, 64 ch.
// Memory/atomic-bound: aggregation scatter dominates (agg fits in 192MB L2).
// GEMMs done in full f32 precision with V_WMMA_F32_16X16X4_F32.

#define NN 100000
#define NE 1000000
#define CH 64
#define TP 68   // padded LDS tile row stride (floats): bank = (4m+k)%64, conflict-free

typedef float v2f __attribute__((ext_vector_type(2)));
typedef float v8f __attribute__((ext_vector_type(8)));

__global__ void zero_kernel(float* __restrict__ p, int n) {
    int i = blockIdx.x * blockDim.x + threadIdx.x;
    int stride = gridDim.x * blockDim.x;
    for (; i < n; i += stride) p[i] = 0.0f;
}

// 16 lanes per edge, float4 per lane: gather feat[src] (coalesced 256B/edge),
// atomic scatter-add to agg[dst]; lane 0 bumps the degree count.
__global__ void scatter_kernel(const float* __restrict__ feat,
                               const int* __restrict__ ei,
                               float* __restrict__ agg,
                               float* __restrict__ cnt,
                               int do_count) {
    int gid = blockIdx.x * blockDim.x + threadIdx.x;   // up to 16M, fits int
    int e  = gid >> 4;
    int c4 = gid & 15;
    if (e >= NE) return;
    int src = ei[e];
    int dst = ei[NE + e];
    float4 v = *(const float4*)(feat + (size_t)src * CH + c4 * 4);
    float* a = agg + (size_t)dst * CH + c4 * 4;
    atomicAdd(a + 0, v.x);
    atomicAdd(a + 1, v.y);
    atomicAdd(a + 2, v.z);
    atomicAdd(a + 3, v.w);
    if (do_count && c4 == 0) atomicAdd(cnt + dst, 1.0f);
}

// One wave per 16-node tile. out = relu(mean @ Wl^T + b + self @ Wr^T).
// FUSE=false: write 16x64 features. FUSE=true: fuse final linear+sigmoid -> yout[N].
template <bool FUSE>
__global__ __launch_bounds__(128)
void sage_conv_kernel(const float* __restrict__ xin,
                      const float* __restrict__ agg,
                      const float* __restrict__ cnt,
                      const float* __restrict__ wl,
                      const float* __restrict__ bias,
                      const float* __restrict__ wr,
                      float* __restrict__ hout,
                      const float* __restrict__ wout,
                      const float* __restrict__ bout,
                      float* __restrict__ yout,
                      int nTiles) {
    __shared__ float smem[4 * 2 * 16 * TP];   // 4 waves x (mean,self) tiles
    const int wave = threadIdx.x >> 5;
    const int lane = threadIdx.x & 31;
    const int tile = blockIdx.x * 4 + wave;
    if (tile >= nTiles) return;               // wave-uniform: EXEC stays all-1s

    float* sMean = smem + wave * (2 * 16 * TP);
    float* sSelf = sMean + 16 * TP;

    const int base = tile * 16;
    const int half = lane >> 4;   // 0 | 1
    const int l16  = lane & 15;

    // Stage self tile and mean tile (agg / max(cnt,1)) into LDS, coalesced.
    #pragma unroll
    for (int i = 0; i < 8; ++i) {
        int rl = half + i * 2;
        int gr = base + rl;
        if (gr >= NN) gr = NN - 1;            // clamp (stores guarded later)
        float4 vs = *(const float4*)(xin + (size_t)gr * CH + l16 * 4);
        *(float4*)(sSelf + rl * TP + l16 * 4) = vs;
        float4 va = *(const float4*)(agg + (size_t)gr * CH + l16 * 4);
        float inv = 1.0f / fmaxf(cnt[gr], 1.0f);
        va.x *= inv; va.y *= inv; va.z *= inv; va.w *= inv;
        *(float4*)(sMean + rl * TP + l16 * 4) = va;
    }
    // wave-private LDS region: no barrier needed (DScnt waits inserted by compiler)

    // ISA 7.12.2 layouts: A 16x4 f32: lanes0-15 K=koff..koff+1, lanes16-31 +2.
    const int koff = half * 2;
    const int m = l16;            // A row for this lane
    const int n = l16;            // B/C/D column for this lane

    float pd[8];
    if (FUSE) {
        #pragma unroll
        for (int r = 0; r < 8; ++r) pd[r] = 0.0f;
    }

    for (int jt = 0; jt < 4; ++jt) {
        const int j0 = jt * 16;
        const float bval = bias[j0 + n];      // bias per output column, all rows
        v8f acc;
        #pragma unroll
        for (int r = 0; r < 8; ++r) acc[r] = bval;

        // B[k][n] = W[(j0+n)*64 + k]  (out = in @ W^T)
        const float* wlrow = wl + (size_t)(j0 + n) * CH + koff;
        const float* wrrow = wr + (size_t)(j0 + n) * CH + koff;

        #pragma unroll
        for (int t = 0; t < 16; ++t) {        // K = 64 = 16 * 4
            const int kb = t * 4;
            v2f am, as2, bl, br;
            am.x  = sMean[m * TP + kb + koff];
            am.y  = sMean[m * TP + kb + koff + 1];
            as2.x = sSelf[m * TP + kb + koff];
            as2.y = sSelf[m * TP + kb + koff + 1];
            bl.x = wlrow[kb];  bl.y = wlrow[kb + 1];
            br.x = wrrow[kb];  br.y = wrrow[kb + 1];
            acc = __builtin_amdgcn_wmma_f32_16x16x4_f32(
                false, am,  false, bl, (short)0, acc, false, false);
            acc = __builtin_amdgcn_wmma_f32_16x16x4_f32(
                false, as2, false, br, (short)0, acc, false, false);
        }

        if (FUSE) {
            const float wo = wout[j0 + n];
            #pragma unroll
            for (int r = 0; r < 8; ++r) pd[r] += fmaxf(acc[r], 0.0f) * wo;
        } else {
            #pragma unroll
            for (int r = 0; r < 8; ++r) {
                int gm = base + r + half * 8;  // C/D: lanes0-15 M=r, lanes16-31 M=r+8
                if (gm < NN) hout[(size_t)gm * CH + j0 + n] = fmaxf(acc[r], 0.0f);
            }
        }
    }

    if (FUSE) {
        const float bo = bout[0];
        #pragma unroll
        for (int r = 0; r < 8; ++r) {
            float v = pd[r];
            // reduce over the 16 N-lanes within each half-wave
            v += __shfl_xor(v, 1, 32);
            v += __shfl_xor(v, 2, 32);
            v += __shfl_xor(v, 4, 32);
            v += __shfl_xor(v, 8, 32);
            if (l16 == 0) {
                int gm = base + r + half * 8;
                if (gm < NN) yout[gm] = 1.0f / (1.0f + expf(-(v + bo)));
            }
        }
    }
}

extern "C" void kernel_launch(void* const* d_in, const int* in_sizes, int n_in,
                              void* d_out, int out_size, void* d_ws, size_t ws_size,
                              hipStream_t stream) {
    const float* x    = (const float*)d_in[0];
    const int*   ei   = (const int*)  d_in[1];
    const float* w1l  = (const float*)d_in[2];
    const float* b1   = (const float*)d_in[3];
    const float* w1r  = (const float*)d_in[4];
    const float* w2l  = (const float*)d_in[5];
    const float* b2   = (const float*)d_in[6];
    const float* w2r  = (const float*)d_in[7];
    const float* wout = (const float*)d_in[8];
    const float* bout = (const float*)d_in[9];
    float* out = (float*)d_out;

    // workspace: agg[N*64] | cnt[N] | h1[N*64]  => ~51.6 MB
    float* agg = (float*)d_ws;
    float* cnt = agg + (size_t)NN * CH;
    float* h1  = cnt + NN;

    const int nTiles = (NN + 15) / 16;                    // 6250
    const int convBlocks = (nTiles + 3) / 4;              // 4 waves/block
    const int scatBlocks = (NE * 16) / 256;               // 62500

    // layer 1
    zero_kernel<<<2048, 256, 0, stream>>>(agg, NN * CH + NN);
    scatter_kernel<<<scatBlocks, 256, 0, stream>>>(x, ei, agg, cnt, 1);
    sage_conv_kernel<false><<<convBlocks, 128, 0, stream>>>(
        x, agg, cnt, w1l, b1, w1r, h1, nullptr, nullptr, nullptr, nTiles);

    // layer 2 (degree counts unchanged; reuse cnt) + fused final linear/sigmoid
    zero_kernel<<<2048, 256, 0, stream>>>(agg, NN * CH);
    scatter_kernel<<<scatBlocks, 256, 0, stream>>>(h1, ei, agg, cnt, 0);
    sage_conv_kernel<true><<<convBlocks, 128, 0, stream>>>(
        h1, agg, cnt, w2l, b2, w2r, nullptr, wout, bout, out, nTiles);
}